// Conv1DLSTMActorModel_85641647882262
// MI455X (gfx1250) — compile-verified
//
#include <hip/hip_runtime.h>
#include <hip/hip_bf16.h>
#include <math.h>

typedef float v2f __attribute__((ext_vector_type(2)));
typedef float v8f __attribute__((ext_vector_type(8)));

#define Bz   128
#define Tt   2048
#define T1   2047      // valid conv output length
#define FF   64        // conv filters
#define HH   100       // LSTM units
#define NG   400       // 4*H gate width  (= 25 tiles of 16)
#define ROWS (Bz * T1) // 262016 = 16376 row-tiles of 16

// ---------------------------------------------------------------------------
// xz = A @ W + bias   (A is either fused-conv(state) [rows,64] or h1seq [rows,100])
// One wave computes one 16x16 C tile with V_WMMA_F32_16X16X4_F32, K/4 steps.
// grid = (2047, 25), block = 256 (8 waves); rowTile = bx*8+wave, colTile = by.
// ---------------------------------------------------------------------------
template <int MODE> // 0: conv-fused from state input (K=64); 1: dense A [rows,100] (K=100)
__global__ void xz_gemm_kernel(const float* __restrict__ SorA,
                               const float* __restrict__ cw,   // conv_w [2,1,64] (MODE 0)
                               const float* __restrict__ cb,   // conv_b [64]     (MODE 0)
                               const float* __restrict__ W,    // [K, 400]
                               const float* __restrict__ bias, // [400]
                               float* __restrict__ xz)         // [rows, 400]
{
    constexpr int K = (MODE == 0) ? FF : HH;
    const int lane    = threadIdx.x & 31;
    const int wave    = threadIdx.x >> 5;
    const int rowTile = blockIdx.x * 8 + wave;  // 0..16375
    const int n0      = blockIdx.y * 16;        // 0..384
    const int mlo     = lane & 15;
    const int hi      = lane >> 4;              // 0 or 1
    const int khalf   = hi * 2;
    const int n       = n0 + mlo;               // B/C column owned by this lane
    const int r       = rowTile * 16 + mlo;     // A row owned by this lane

    v8f acc = {0.f, 0.f, 0.f, 0.f, 0.f, 0.f, 0.f, 0.f};

    if constexpr (MODE == 0) {
        // conv-on-the-fly: x[r,k] = relu(s[b,t]*cw[0,k] + s[b,t+1]*cw[1,k] + cb[k])
        const unsigned b = (unsigned)r / T1;
        const unsigned t = (unsigned)r - b * T1;
        const float s0 = SorA[b * Tt + t];
        const float s1 = SorA[b * Tt + t + 1];
#pragma unroll
        for (int ks = 0; ks < K / 4; ++ks) {
            const int kb = ks * 4 + khalf;
            v2f a, bb;
            a.x  = fmaxf(fmaf(s1, cw[FF + kb],     fmaf(s0, cw[kb],     cb[kb])),     0.f);
            a.y  = fmaxf(fmaf(s1, cw[FF + kb + 1], fmaf(s0, cw[kb + 1], cb[kb + 1])), 0.f);
            bb.x = W[(size_t)kb * NG + n];
            bb.y = W[(size_t)(kb + 1) * NG + n];
            acc = __builtin_amdgcn_wmma_f32_16x16x4_f32(false, a, false, bb,
                                                        (short)0, acc, false, false);
        }
    } else {
        const float* A = SorA + (size_t)r * HH;
#pragma unroll
        for (int ks = 0; ks < K / 4; ++ks) {
            const int kb = ks * 4 + khalf;
            v2f a, bb;
            a.x  = A[kb];
            a.y  = A[kb + 1];
            bb.x = W[(size_t)kb * NG + n];
            bb.y = W[(size_t)(kb + 1) * NG + n];
            acc = __builtin_amdgcn_wmma_f32_16x16x4_f32(false, a, false, bb,
                                                        (short)0, acc, false, false);
        }
    }

    const float bs = bias[n];
    float* out = xz + (size_t)rowTile * 16 * NG + n;
#pragma unroll
    for (int j = 0; j < 8; ++j) {
        const int m = j + 8 * hi;               // C/D layout: VGPR j -> M = j (+8 for hi lanes)
        out[(size_t)m * NG] = acc[j] + bs;
    }
}

// ---------------------------------------------------------------------------
// LSTM scan. grid = 8 (batch tiles of 16 rows), block = 800 (25 waves).
// Wave w owns gate columns [16w,16w+16). U fragments live in 50 VGPRs per wave.
// h,c,z in LDS. z_t = xz[:,t,:] (bias pre-folded) + h_{t-1} @ U via 25 f32 WMMAs.
// ---------------------------------------------------------------------------
__global__ void lstm_scan_kernel(const float* __restrict__ xz, // [B*T1, 400]
                                 const float* __restrict__ U,  // [100, 400]
                                 float* __restrict__ hout,     // seq [B*T1,100] or final [B,100]
                                 int storeSeq)
{
    __shared__ float hs[16][HH + 1];
    __shared__ float cs[16][HH + 1];
    __shared__ float zs[16][NG];

    const int tid  = threadIdx.x;   // 0..799
    const int lane = tid & 31;
    const int wave = tid >> 5;      // 0..24 == column tile
    const int b0   = blockIdx.x * 16;

    for (int i = tid; i < 16 * (HH + 1); i += 800) {
        (&hs[0][0])[i] = 0.f;
        (&cs[0][0])[i] = 0.f;
    }
    __syncthreads();

    const int mlo = lane & 15;
    const int hi  = lane >> 4;
    const int khalf = hi * 2;
    const int n0  = wave * 16;
    const int n   = n0 + mlo;

    // Register-resident B fragments of U for this wave's 16 columns (time-invariant).
    v2f bfrag[25];
#pragma unroll
    for (int ks = 0; ks < 25; ++ks) {
        const int kb = ks * 4 + khalf;
        bfrag[ks].x = U[(size_t)kb * NG + n];
        bfrag[ks].y = U[(size_t)(kb + 1) * NG + n];
    }

    // Each thread owns 2 of the 1600 (m,j) gate elements.
    const int m0i = tid / 100,          j0i = tid - m0i * 100;
    const int m1i = (tid + 800) / 100,  j1i = (tid + 800) - m1i * 100;

    for (int t = 0; t < T1; ++t) {
        // C <- xz tile for this (batch tile, t, column tile)
        const float* xzp = xz + ((size_t)b0 * T1 + t) * NG;
        v8f acc;
#pragma unroll
        for (int j = 0; j < 8; ++j) {
            const int m = j + 8 * hi;
            acc[j] = xzp[(size_t)m * T1 * NG + n];
        }
        if (t + 1 < T1) { // prefetch next timestep's xz row for one of our 8 rows
            __builtin_prefetch(xzp + NG + (size_t)(8 * hi) * T1 * NG + n, 0, 0);
        }

        // acc += h_{t-1} @ U[:, cols]   (K = 100 = 25 x 4)
#pragma unroll
        for (int ks = 0; ks < 25; ++ks) {
            const int kb = ks * 4 + khalf;
            v2f a;
            a.x = hs[mlo][kb];
            a.y = hs[mlo][kb + 1];
            acc = __builtin_amdgcn_wmma_f32_16x16x4_f32(false, a, false, bfrag[ks],
                                                        (short)0, acc, false, false);
        }

        // publish z tile
#pragma unroll
        for (int j = 0; j < 8; ++j) {
            const int m = j + 8 * hi;
            zs[m][n] = acc[j];
        }
        __syncthreads();

        // gate math: i,f,o = sigmoid; g = relu; c = f*c + i*g; h = o*relu(c)
#pragma unroll
        for (int e = 0; e < 2; ++e) {
            const int m = e ? m1i : m0i;
            const int j = e ? j1i : j0i;
            const float zi = zs[m][j];
            const float zf = zs[m][HH + j];
            const float zg = zs[m][2 * HH + j];
            const float zo = zs[m][3 * HH + j];
            const float ig = 1.f / (1.f + __expf(-zi));
            const float fg = 1.f / (1.f + __expf(-zf));
            const float gg = fmaxf(zg, 0.f);
            const float og = 1.f / (1.f + __expf(-zo));
            const float c  = fg * cs[m][j] + ig * gg;
            const float h  = og * fmaxf(c, 0.f);
            cs[m][j] = c;
            hs[m][j] = h;
            if (storeSeq) {
                hout[((size_t)(b0 + m) * T1 + t) * HH + j] = h;
            } else if (t == T1 - 1) {
                hout[(size_t)(b0 + m) * HH + j] = h;
            }
        }
        __syncthreads();
    }
}

// ---------------------------------------------------------------------------
// logits = h2 @ Wd + bd ; softmax over A=3. One thread per batch row.
// ---------------------------------------------------------------------------
__global__ void dense_softmax_kernel(const float* __restrict__ h2,
                                     const float* __restrict__ Wd, // [100,3]
                                     const float* __restrict__ bd, // [3]
                                     float* __restrict__ out)      // [128,3]
{
    const int b = blockIdx.x * blockDim.x + threadIdx.x;
    if (b >= Bz) return;
    float a0 = bd[0], a1 = bd[1], a2 = bd[2];
    const float* h = h2 + (size_t)b * HH;
#pragma unroll 4
    for (int j = 0; j < HH; ++j) {
        const float hv = h[j];
        a0 = fmaf(hv, Wd[j * 3 + 0], a0);
        a1 = fmaf(hv, Wd[j * 3 + 1], a1);
        a2 = fmaf(hv, Wd[j * 3 + 2], a2);
    }
    const float mx = fmaxf(a0, fmaxf(a1, a2));
    const float e0 = __expf(a0 - mx), e1 = __expf(a1 - mx), e2 = __expf(a2 - mx);
    const float inv = 1.f / (e0 + e1 + e2);
    out[b * 3 + 0] = e0 * inv;
    out[b * 3 + 1] = e1 * inv;
    out[b * 3 + 2] = e2 * inv;
}

// ---------------------------------------------------------------------------
extern "C" void kernel_launch(void* const* d_in, const int* in_sizes, int n_in,
                              void* d_out, int out_size, void* d_ws, size_t ws_size,
                              hipStream_t stream) {
    const float* state   = (const float*)d_in[0];  // [128,2048,1]
    const float* conv_w  = (const float*)d_in[1];  // [2,1,64]
    const float* conv_b  = (const float*)d_in[2];  // [64]
    const float* lstm1_w = (const float*)d_in[3];  // [64,400]
    const float* lstm1_u = (const float*)d_in[4];  // [100,400]
    const float* lstm1_b = (const float*)d_in[5];  // [400]
    const float* lstm2_w = (const float*)d_in[6];  // [100,400]
    const float* lstm2_u = (const float*)d_in[7];  // [100,400]
    const float* lstm2_b = (const float*)d_in[8];  // [400]
    const float* dense_w = (const float*)d_in[9];  // [100,3]
    const float* dense_b = (const float*)d_in[10]; // [3]
    float* out = (float*)d_out;

    // workspace layout
    float* xz    = (float*)d_ws;                       // [262016, 400] (reused by both layers)
    float* h1seq = xz + (size_t)ROWS * NG;             // [262016, 100]
    float* h2    = h1seq + (size_t)ROWS * HH;          // [128, 100]

    const dim3 ggrid(T1, NG / 16);  // 2047 x 25; 8 row-tiles per block

    // Layer 1 input projection (conv fused) + bias
    xz_gemm_kernel<0><<<ggrid, 256, 0, stream>>>(state, conv_w, conv_b,
                                                 lstm1_w, lstm1_b, xz);
    // Layer 1 recurrent scan, store full h sequence
    lstm_scan_kernel<<<Bz / 16, 800, 0, stream>>>(xz, lstm1_u, h1seq, 1);
    // Layer 2 input projection + bias
    xz_gemm_kernel<1><<<ggrid, 256, 0, stream>>>(h1seq, nullptr, nullptr,
                                                 lstm2_w, lstm2_b, xz);
    // Layer 2 recurrent scan, store only final h
    lstm_scan_kernel<<<Bz / 16, 800, 0, stream>>>(xz, lstm2_u, h2, 0);
    // Dense + softmax
    dense_softmax_kernel<<<1, 128, 0, stream>>>(h2, dense_w, dense_b, out);
}